// MRConv2d_22333829939359
// MI455X (gfx1250) — compile-verified
//
#include <hip/hip_runtime.h>
#include <hip/hip_bf16.h>
#include <stdint.h>

// ---------------------------------------------------------------------------
// MRConv2d (max-relative graph conv) for MI455X / gfx1250, wave32.
//   B=4, C=192, N=4096, E=262144, C_OUT=384, BN=16384.
// Pipeline:
//   1) prep_kernel : transpose x (B,C,N) -> cat[node][2c] and init cat[node][2c+1]=-inf
//   2) edge_kernel : per-edge atomic scatter-max of (x[src]-x[dst]) into odd channels
//   3) gemm_kernel : out = ReLU(W @ cat + b) via V_WMMA_F32_16X16X4_F32 (exact fp32),
//                    register-blocked 64x32 per wave (8 WMMAs per k-step, 6 loads).
// ---------------------------------------------------------------------------

typedef __attribute__((ext_vector_type(2))) float v2f;
typedef __attribute__((ext_vector_type(4))) float v4f;
typedef __attribute__((ext_vector_type(8))) float v8f;

#define Bm   4
#define Cm   192
#define Nm   4096
#define BN   (Bm * Nm)       // 16384 nodes
#define C2   (2 * Cm)        // 384 (= GEMM K)
#define COUT 384             // GEMM N
#define NEG_INF __builtin_inff()

#define TM 4                 // 16-row subtiles per wave  (64 rows)
#define TO 2                 // 16-col subtiles per wave  (32 cols)

// -------------------------------------------------------------------
// Kernel 1: build interleaved node table cat[node][2c]=x, [2c+1]=-inf
// Coalesced 8B stores; strided reads are a single HBM pass (12.6 MB).
// -------------------------------------------------------------------
__global__ __launch_bounds__(256) void prep_kernel(const float* __restrict__ x,
                                                   float* __restrict__ cat) {
  int idx = blockIdx.x * 256 + threadIdx.x;       // over BN*C = 3,145,728
  if (idx >= BN * Cm) return;
  int c    = idx % Cm;
  int node = idx / Cm;
  int b    = node >> 12;                          // node / 4096
  int n    = node & 4095;
  float v = x[((size_t)b * Cm + c) * Nm + n];
  v2f pair;
  pair.x = v;
  pair.y = -NEG_INF;                              // -inf sentinel for segment_max
  *(v2f*)(cat + (size_t)node * C2 + 2 * c) = pair;
}

// -------------------------------------------------------------------
// Float atomic-max via monotone integer reinterpretation:
//   val >= 0 : signed   atomicMax (positive float bits are ordered ints)
//   val <  0 : unsigned atomicMin (negative float bits reverse-ordered)
// Canonicalize -0 -> +0 so the sign split stays monotone.
// Lowers to global_atomic_max_i32 / global_atomic_min_u32 (no CAS loop).
// -------------------------------------------------------------------
__device__ __forceinline__ void atomic_max_f32(float* addr, float val) {
  val = (val == 0.0f) ? 0.0f : val;
  if (val >= 0.0f) atomicMax((int*)addr, __float_as_int(val));
  else             atomicMin((unsigned int*)addr, __float_as_uint(val));
}

// -------------------------------------------------------------------
// Kernel 2: one wave per edge; 192 channels in 6 strips of 32 lanes.
// Reads are 8B/lane coalesced float2 (x lives at even offsets);
// atomics land on the odd offsets of the dst row. All traffic L2-resident.
// -------------------------------------------------------------------
__global__ __launch_bounds__(256) void edge_kernel(float* __restrict__ cat,
                                                   const int* __restrict__ ei,
                                                   int E) {
  int wave = threadIdx.x >> 5;
  int lane = threadIdx.x & 31;
  int e = blockIdx.x * 8 + wave;
  if (e >= E) return;
  int dst = ei[e];        // edge_index[0][e]
  int src = ei[E + e];    // edge_index[1][e]
  const v2f* drow = (const v2f*)(cat + (size_t)dst * C2);
  const v2f* srow = (const v2f*)(cat + (size_t)src * C2);
  float* dagg = cat + (size_t)dst * C2;
#pragma unroll
  for (int c = lane; c < Cm; c += 32) {
    float xd = drow[c].x;
    float xs = srow[c].x;
    atomic_max_f32(dagg + 2 * c + 1, xs - xd);
  }
}

// -------------------------------------------------------------------
// Kernel 3: fp32 GEMM + bias + ReLU with V_WMMA_F32_16X16X4_F32.
// Register-blocked: each wave owns a 64x32 output tile = TM x TO subtiles.
//   per k-step (K block of 4): TM A-loads + TO B-loads -> TM*TO WMMAs
//   A[m][k] = cat[m0+m][k]  (with -inf -> 0 folded into the load)
//   B[k][n] = W[o0+n][k]    (W row-major (COUT, 2C) -> B = W^T tile)
// Lane layout (ISA 7.12.2): lr=lane&15, half=lane>>4.
//   A/B frag: vgpr0,1 hold K = 2*half, 2*half+1 for row/col lr -> one b64 load.
//   C/D: vgpr j holds row m0+8*half+j, col o0+lr -> contiguous-n float4 stores.
// -------------------------------------------------------------------
__global__ __launch_bounds__(256) void gemm_kernel(const float* __restrict__ cat,
                                                   const float* __restrict__ W,
                                                   const float* __restrict__ bias,
                                                   float* __restrict__ out) {
  const int wave = threadIdx.x >> 5;
  const int lane = threadIdx.x & 31;
  const int tile = blockIdx.x * 8 + wave;          // 3072 tiles total
  const int OT   = COUT / (16 * TO);               // 12 o-blocks
  const int tm   = tile / OT;                      // 0..255
  const int to   = tile % OT;
  const int m0   = tm * 16 * TM;                   // 64-row block (within one b)
  const int o0   = to * 16 * TO;                   // 32-col block
  const int half = lane >> 4;
  const int lr   = lane & 15;

  const float* ap[TM];
#pragma unroll
  for (int i = 0; i < TM; ++i)
    ap[i] = cat + (size_t)(m0 + 16 * i + lr) * C2 + 2 * half;
  const float* bp[TO];
#pragma unroll
  for (int j = 0; j < TO; ++j)
    bp[j] = W + (size_t)(o0 + 16 * j + lr) * C2 + 2 * half;

  v8f acc[TM][TO];
#pragma unroll
  for (int i = 0; i < TM; ++i)
#pragma unroll
    for (int j = 0; j < TO; ++j) acc[i][j] = (v8f){};

#pragma unroll 2
  for (int k0 = 0; k0 < C2; k0 += 4) {
    v2f a[TM], b[TO];
#pragma unroll
    for (int i = 0; i < TM; ++i) {
      a[i] = *(const v2f*)(ap[i] + k0);
      // fold segment_max empty-segment fixup (-inf -> 0) into the A load
      a[i].x = (a[i].x == -NEG_INF) ? 0.0f : a[i].x;
      a[i].y = (a[i].y == -NEG_INF) ? 0.0f : a[i].y;
    }
#pragma unroll
    for (int j = 0; j < TO; ++j) b[j] = *(const v2f*)(bp[j] + k0);
#pragma unroll
    for (int i = 0; i < TM; ++i)
#pragma unroll
      for (int j = 0; j < TO; ++j)
        // (neg_a, A, neg_b, B, c_mod, C, reuse_a, reuse_b)
        acc[i][j] = __builtin_amdgcn_wmma_f32_16x16x4_f32(
            false, a[i], false, b[j], (short)0, acc[i][j], false, false);
  }

  // epilogue: bias + ReLU, two aligned float4 stores per subtile
#pragma unroll
  for (int i = 0; i < TM; ++i) {
    const int mrow  = m0 + 16 * i + 8 * half;      // first of 8 consecutive rows
    const int b_idx = mrow >> 12;
    const int nb    = mrow & 4095;
#pragma unroll
    for (int j = 0; j < TO; ++j) {
      const float bv = bias[o0 + 16 * j + lr];
      float* op = out + (size_t)b_idx * COUT * Nm
                      + (size_t)(o0 + 16 * j + lr) * Nm + nb;
      v4f lo, hi;
#pragma unroll
      for (int q = 0; q < 4; ++q) lo[q] = fmaxf(acc[i][j][q] + bv, 0.0f);
#pragma unroll
      for (int q = 0; q < 4; ++q) hi[q] = fmaxf(acc[i][j][4 + q] + bv, 0.0f);
      *(v4f*)(op)     = lo;   // rows m..m+3  == contiguous n
      *(v4f*)(op + 4) = hi;   // rows m+4..m+7
    }
  }
}

// ---------------------------------------------------------------------------
extern "C" void kernel_launch(void* const* d_in, const int* in_sizes, int n_in,
                              void* d_out, int out_size, void* d_ws, size_t ws_size,
                              hipStream_t stream) {
  const float* x    = (const float*)d_in[0];   // (B, C, N, 1) fp32
  const int*   ei   = (const int*)d_in[1];     // (2, E)
  const float* W    = (const float*)d_in[2];   // (COUT, 2C) fp32
  const float* bias = (const float*)d_in[3];   // (COUT,)
  float*       out  = (float*)d_out;           // (B, COUT, N, 1)
  float*       cat  = (float*)d_ws;            // (BN, 2C) fp32 = 25.2 MB

  const int E = in_sizes[1] / 2;

  prep_kernel<<<(BN * Cm + 255) / 256, 256, 0, stream>>>(x, cat);
  edge_kernel<<<(E + 7) / 8, 256, 0, stream>>>(cat, ei, E);

  const int tiles = (BN / (16 * TM)) * (COUT / (16 * TO));   // 3072
  gemm_kernel<<<tiles / 8, 256, 0, stream>>>(cat, W, bias, out);
}